// SoftmaxFreeAttentionKernel_81698867904920
// MI455X (gfx1250) — compile-verified
//
#include <hip/hip_runtime.h>
#include <math.h>

// ---------------------------------------------------------------------------
// Softmax-free (Nystrom/Gaussian-kernel) attention for MI455X (gfx1250).
// All matrix math via V_WMMA_F32_16X16X4_F32 (fp32-exact, matches reference).
// Newton-Schulz working set (~103 MB) is L2-resident (192 MB L2).
// GEMM register-blocked 32x32 per wave: 2 dwords loaded per WMMA from L2.
// ---------------------------------------------------------------------------

typedef __attribute__((ext_vector_type(2))) float v2f;
typedef __attribute__((ext_vector_type(8))) float v8f;

#define BATCH  16
#define NHEAD  8
#define BH     128          // BATCH * NHEAD
#define HW     56           // H = W
#define NSEQ   3136         // 56*56
#define HD     64
#define MLAND  196          // 14*14 landmarks
#define MP     224          // landmarks padded to 14*16 (even 32-blocks)
#define MT     14           // landmark tiles (16-wide)
#define MB     7            // landmark 32-blocks
#define NT     196          // sequence tiles (3136/16)
#define QSCALE 0.35355339059327373f  // 64^(-1/4)
#define LNEPS  1e-5f
#define NEWTON_ITERS 20

__device__ __forceinline__ v8f wmma4(v2f a, v2f b, v8f c) {
  // D = A(16x4,f32) * B(4x16,f32) + C(16x16,f32); neg/reuse mods off.
  return __builtin_amdgcn_wmma_f32_16x16x4_f32(false, a, false, b, (short)0, c,
                                               false, false);
}

// A-fragment from row-major M[ld]: lane holds row m0+(lane&15), k = k0+(lane>>4)*2.
// Also used to build B = Xᵀ fragments (identical per-lane load pattern).
__device__ __forceinline__ v2f fragA(const float* __restrict__ M, int ld,
                                     int m0, int k0, int lane) {
  const float* p = M + (size_t)(m0 + (lane & 15)) * ld + (k0 + ((lane >> 4) << 1));
  v2f r; r.x = p[0]; r.y = p[1];
  return r;
}

// B-fragment from row-major B'[K][N]: lane holds col n0+(lane&15), k rows k..k+1.
__device__ __forceinline__ v2f fragB(const float* __restrict__ M, int ld,
                                     int k0, int n0, int lane) {
  int k = k0 + ((lane >> 4) << 1);
  int n = n0 + (lane & 15);
  v2f r; r.x = M[(size_t)k * ld + n]; r.y = M[(size_t)(k + 1) * ld + n];
  return r;
}

// --------------------------- Stage 1: landmarks -----------------------------
// Ql = GELU(LN(conv4x4s4(Q*QSCALE))), plus 0.5*||ql||^2 (pad rows -> 1e30).
__global__ void __launch_bounds__(256) landmark_kernel(
    const float* __restrict__ Q, const float* __restrict__ w_land,
    const float* __restrict__ gamma, const float* __restrict__ beta,
    float* __restrict__ QL, float* __restrict__ QLN) {
  __shared__ float raw[MLAND * HD];  // 50 KB of the 320 KB WGP LDS
  int bh = blockIdx.x, tid = threadIdx.x;
  const float* Qb = Q + (size_t)bh * NSEQ * HD;
  for (int o = tid; o < MLAND * HD; o += 256) {
    int l = o >> 6, c = o & 63;
    int ty = l / 14, tx = l - 14 * (l / 14);
    float acc = 0.f;
    for (int dy = 0; dy < 4; ++dy)
      for (int dx = 0; dx < 4; ++dx) {
        const float* qp = Qb + (size_t)((ty * 4 + dy) * HW + tx * 4 + dx) * HD;
        const float* wp = w_land + (size_t)((dy * 4 + dx) * HD) * HD + c;
        for (int i = 0; i < HD; ++i) acc += qp[i] * wp[(size_t)i * HD];
      }
    raw[o] = acc * QSCALE;  // conv is linear: fold Q scaling in afterwards
  }
  __syncthreads();
  for (int l = tid; l < MP; l += 256) {
    float* out = QL + ((size_t)bh * MP + l) * HD;
    if (l < MLAND) {
      const float* r = raw + l * HD;
      float mu = 0.f;
      for (int c = 0; c < HD; ++c) mu += r[c];
      mu *= (1.0f / HD);
      float var = 0.f;
      for (int c = 0; c < HD; ++c) { float d = r[c] - mu; var += d * d; }
      var *= (1.0f / HD);
      float rstd = rsqrtf(var + LNEPS);
      float nrm = 0.f;
      for (int c = 0; c < HD; ++c) {
        float v = (r[c] - mu) * rstd * gamma[c] + beta[c];
        float g = 0.5f * v * (1.0f + erff(v * 0.70710678118654752f));  // exact GELU
        out[c] = g;
        nrm += g * g;
      }
      QLN[(size_t)bh * MP + l] = 0.5f * nrm;
    } else {  // zero pad rows; huge norm => exp(...) == 0 for all padded entries
      for (int c = 0; c < HD; ++c) out[c] = 0.f;
      QLN[(size_t)bh * MP + l] = 1e30f;
    }
  }
}

// 0.5*||QSCALE*q_i||^2
__global__ void __launch_bounds__(256) qnorm_kernel(const float* __restrict__ Q,
                                                    float* __restrict__ QN) {
  int idx = blockIdx.x * 256 + threadIdx.x;
  if (idx >= BH * NSEQ) return;
  const float* q = Q + (size_t)idx * HD;
  float s = 0.f;
  for (int c = 0; c < HD; ++c) { float v = q[c] * QSCALE; s += v * v; }
  QN[idx] = 0.5f * s;
}

// ----------------------- Stage 2: k2 Gram via WMMA --------------------------
// P[j][i] = exp(ql_j . ql_i - 0.5||ql_j||^2 - 0.5||ql_i||^2); padded -> 0.
__global__ void __launch_bounds__(32) k2_kernel(const float* __restrict__ QL,
                                                const float* __restrict__ QLN,
                                                float* __restrict__ P) {
  int lane = threadIdx.x;
  int bh = blockIdx.z, j0 = blockIdx.y * 16, i0 = blockIdx.x * 16;
  const float* Qlb = QL + (size_t)bh * MP * HD;
  const float* qlnb = QLN + (size_t)bh * MP;
  v8f acc = {};
#pragma unroll
  for (int k0 = 0; k0 < HD; k0 += 4) {
    v2f a = fragA(Qlb, HD, j0, k0, lane);
    v2f b = fragA(Qlb, HD, i0, k0, lane);  // B = Qlᵀ: same per-lane pattern
    acc = wmma4(a, b, acc);
  }
  int n = i0 + (lane & 15);
  int hi8 = (lane >> 4) << 3;
  float bn = qlnb[n];
  float* Pb = P + (size_t)bh * MP * MP;
#pragma unroll
  for (int r = 0; r < 8; ++r) {
    int m = j0 + r + hi8;
    Pb[(size_t)m * MP + n] = __expf(acc[r] - qlnb[m] - bn);
  }
}

// V0 = Pᵀ / (max_col_sum * max_row_sum); pad rows/cols stay exactly zero.
__global__ void __launch_bounds__(256) newton_init_kernel(
    const float* __restrict__ P, float* __restrict__ V0) {
  __shared__ float red[256];
  int bh = blockIdx.x, tid = threadIdx.x;
  const float* Pb = P + (size_t)bh * MP * MP;
  float local = 0.f;
  for (int i = tid; i < MP; i += 256) {
    float s = 0.f;
    for (int j = 0; j < MP; ++j) s += fabsf(Pb[(size_t)j * MP + i]);
    local = fmaxf(local, s);
  }
  red[tid] = local; __syncthreads();
  for (int w = 128; w > 0; w >>= 1) {
    if (tid < w) red[tid] = fmaxf(red[tid], red[tid + w]);
    __syncthreads();
  }
  float n1 = red[0]; __syncthreads();
  local = 0.f;
  for (int j = tid; j < MP; j += 256) {
    float s = 0.f;
    const float* row = Pb + (size_t)j * MP;
    for (int i = 0; i < MP; ++i) s += fabsf(row[i]);
    local = fmaxf(local, s);
  }
  red[tid] = local; __syncthreads();
  for (int w = 128; w > 0; w >>= 1) {
    if (tid < w) red[tid] = fmaxf(red[tid], red[tid + w]);
    __syncthreads();
  }
  float rcp = 1.0f / (n1 * red[0]);
  float* Vb = V0 + (size_t)bh * MP * MP;
  for (int idx = tid; idx < MP * MP; idx += 256) {
    int i = idx / MP, j = idx - MP * (idx / MP);
    Vb[idx] = Pb[(size_t)j * MP + i] * rcp;
  }
}

// ------------- Batched fp32 GEMM, 32x32 register-blocked (WMMA) --------------
// Each wave owns a 32x32 output block (2x2 tiles, 4 v8f accumulators).
// Per k-step: 2 A-frags + 2 B-frags (8 dwords/lane) feed 4 independent WMMAs.
// mode 0: C = A@B      mode 1: C = 2*E - A@B   (Newton update, E == current V)
__global__ void __launch_bounds__(32) gemm_wmma_kernel(
    float* __restrict__ C, const float* __restrict__ A,
    const float* __restrict__ B, const float* __restrict__ E, int K, int ldc,
    int lda, int ldb, int lde, long long strC, long long strA, long long strB,
    long long strE, int mode) {
  int lane = threadIdx.x;
  int bh = blockIdx.z, m0 = blockIdx.y * 32, n0 = blockIdx.x * 32;
  const float* Ab = A + (size_t)bh * strA;
  const float* Bb = B + (size_t)bh * strB;
  v8f acc[2][2] = {{{}, {}}, {{}, {}}};
  for (int k0 = 0; k0 < K; k0 += 4) {
    v2f a0 = fragA(Ab, lda, m0, k0, lane);
    v2f a1 = fragA(Ab, lda, m0 + 16, k0, lane);
    v2f b0 = fragB(Bb, ldb, k0, n0, lane);
    v2f b1 = fragB(Bb, ldb, k0, n0 + 16, lane);
    acc[0][0] = wmma4(a0, b0, acc[0][0]);
    acc[0][1] = wmma4(a0, b1, acc[0][1]);
    acc[1][0] = wmma4(a1, b0, acc[1][0]);
    acc[1][1] = wmma4(a1, b1, acc[1][1]);
  }
  int hi8 = (lane >> 4) << 3;
  float* Cb = C + (size_t)bh * strC;
  const float* Eb = E + (size_t)bh * strE;
#pragma unroll
  for (int mi = 0; mi < 2; ++mi) {
#pragma unroll
    for (int ni = 0; ni < 2; ++ni) {
      int n = n0 + ni * 16 + (lane & 15);
#pragma unroll
      for (int r = 0; r < 8; ++r) {
        int m = m0 + mi * 16 + r + hi8;
        float v = acc[mi][ni][r];
        if (mode == 1) v = 2.0f * Eb[(size_t)m * lde + n] - v;
        Cb[(size_t)m * ldc + n] = v;
      }
    }
  }
}

// ---------------- Stage 3: Z = k1ᵀ @ V, fused k1 tiles -----------------------
// A-fragments (Ql rows j0) are invariant across the 196 i-tiles: hoisted.
__global__ void __launch_bounds__(32) zk_kernel(
    const float* __restrict__ Q, const float* __restrict__ V,
    const float* __restrict__ QL, const float* __restrict__ QLN,
    const float* __restrict__ QN, float* __restrict__ Z) {
  __shared__ float S[16][17];
  int lane = threadIdx.x;
  int j0 = blockIdx.x * 16, bh = blockIdx.y;
  const float* Qb  = Q  + (size_t)bh * NSEQ * HD;
  const float* Vb  = V  + (size_t)bh * NSEQ * HD;
  const float* Qlb = QL + (size_t)bh * MP * HD;
  const float* qlnb = QLN + (size_t)bh * MP;
  const float* qnb  = QN  + (size_t)bh * NSEQ;
  int lm = lane & 15, hi8 = (lane >> 4) << 3;
  float qln_r[8];
#pragma unroll
  for (int r = 0; r < 8; ++r) qln_r[r] = qlnb[j0 + r + hi8];
  v2f aQl[16];  // hoisted invariant A-fragments (32 VGPRs)
#pragma unroll
  for (int kk = 0; kk < 16; ++kk) aQl[kk] = fragA(Qlb, HD, j0, kk * 4, lane);
  v8f acc[4] = {{}, {}, {}, {}};
  for (int it = 0; it < NT; ++it) {
    int i0 = it * 16;
    v8f d = {};
#pragma unroll
    for (int kk = 0; kk < 16; ++kk) {
      v2f b = fragA(Qb, HD, i0, kk * 4, lane);  // B = (s*Q)ᵀ
      b.x *= QSCALE; b.y *= QSCALE;
      d = wmma4(aQl[kk], b, d);
    }
    float qn_n = qnb[i0 + lm];
    __syncthreads();
#pragma unroll
    for (int r = 0; r < 8; ++r)
      S[r + hi8][lm] = __expf(d[r] - qln_r[r] - qn_n);  // k1ᵀ tile, D-layout
    __syncthreads();
#pragma unroll
    for (int dt = 0; dt < 4; ++dt) {
      int d0 = dt * 16;
#pragma unroll
      for (int k0 = 0; k0 < 16; k0 += 4) {
        int kk = k0 + ((lane >> 4) << 1);
        v2f a; a.x = S[lm][kk]; a.y = S[lm][kk + 1];
        v2f b = fragB(Vb, HD, i0 + k0, d0, lane);
        acc[dt] = wmma4(a, b, acc[dt]);
      }
    }
  }
  float* Zb = Z + (size_t)bh * MP * HD;
#pragma unroll
  for (int dt = 0; dt < 4; ++dt) {
#pragma unroll
    for (int r = 0; r < 8; ++r)
      Zb[(size_t)(j0 + r + hi8) * HD + dt * 16 + lm] = acc[dt][r];
  }
}

// -------- Stage 4: out = k1 @ Y + depthwise3x3-over-heads(V), fused ---------
// A-fragments (scaled Q rows i0) invariant across the 14 j-tiles: hoisted.
__global__ void __launch_bounds__(32) x_kernel(
    const float* __restrict__ Q, const float* __restrict__ V,
    const float* __restrict__ QL, const float* __restrict__ QLN,
    const float* __restrict__ QN, const float* __restrict__ Y,
    const float* __restrict__ conv_w, float* __restrict__ OUT) {
  __shared__ float S[16][17];
  int lane = threadIdx.x;
  int i0 = blockIdx.x * 16, bh = blockIdx.y;
  const float* Qb  = Q  + (size_t)bh * NSEQ * HD;
  const float* Vb  = V  + (size_t)bh * NSEQ * HD;
  const float* Qlb = QL + (size_t)bh * MP * HD;
  const float* qlnb = QLN + (size_t)bh * MP;
  const float* qnb  = QN  + (size_t)bh * NSEQ;
  const float* Yb   = Y  + (size_t)bh * MP * HD;
  int lm = lane & 15, hi8 = (lane >> 4) << 3;
  float qn_r[8];
#pragma unroll
  for (int r = 0; r < 8; ++r) qn_r[r] = qnb[i0 + r + hi8];
  v2f aQ[16];  // hoisted invariant A-fragments (scaled)
#pragma unroll
  for (int kk = 0; kk < 16; ++kk) {
    v2f t = fragA(Qb, HD, i0, kk * 4, lane);
    t.x *= QSCALE; t.y *= QSCALE;
    aQ[kk] = t;
  }
  v8f acc[4] = {{}, {}, {}, {}};
  for (int jt = 0; jt < MT; ++jt) {
    int j0 = jt * 16;
    v8f d = {};
#pragma unroll
    for (int kk = 0; kk < 16; ++kk) {
      v2f b = fragA(Qlb, HD, j0, kk * 4, lane);  // B = Qlᵀ
      d = wmma4(aQ[kk], b, d);
    }
    float qln_n = qlnb[j0 + lm];
    __syncthreads();
#pragma unroll
    for (int r = 0; r < 8; ++r)
      S[r + hi8][lm] = __expf(d[r] - qn_r[r] - qln_n);  // k1 tile
    __syncthreads();
#pragma unroll
    for (int dt = 0; dt < 4; ++dt) {
      int d0 = dt * 16;
#pragma unroll
      for (int k0 = 0; k0 < 16; k0 += 4) {
        int kk = k0 + ((lane >> 4) << 1);
        v2f a; a.x = S[lm][kk]; a.y = S[lm][kk + 1];
        v2f b = fragB(Yb, HD, j0 + k0, d0, lane);
        acc[dt] = wmma4(a, b, acc[dt]);
      }
    }
  }
  int h = bh & (NHEAD - 1);
  float w[9];
#pragma unroll
  for (int t = 0; t < 9; ++t) w[t] = conv_w[t * NHEAD + h];
#pragma unroll
  for (int dt = 0; dt < 4; ++dt) {
#pragma unroll
    for (int r = 0; r < 8; ++r) {
      int m = i0 + r + hi8;
      int c = dt * 16 + lm;
      int y = m / HW, x = m - y * HW;
      float conv = 0.f;
#pragma unroll
      for (int dy = -1; dy <= 1; ++dy) {
        int yy = y + dy;
        if (yy < 0 || yy >= HW) continue;
#pragma unroll
        for (int dx = -1; dx <= 1; ++dx) {
          int xx = x + dx;
          if (xx < 0 || xx >= HW) continue;
          conv += w[(dy + 1) * 3 + (dx + 1)] * Vb[(size_t)(yy * HW + xx) * HD + c];
        }
      }
      OUT[((size_t)bh * NSEQ + m) * HD + c] = acc[dt][r] + conv;
    }
  }
}

// ---------------------------------------------------------------------------
extern "C" void kernel_launch(void* const* d_in, const int* in_sizes, int n_in,
                              void* d_out, int out_size, void* d_ws,
                              size_t ws_size, hipStream_t stream) {
  (void)in_sizes; (void)n_in; (void)out_size; (void)ws_size;
  const float* Q      = (const float*)d_in[0];
  const float* V      = (const float*)d_in[1];
  const float* w_land = (const float*)d_in[2];
  const float* gamma  = (const float*)d_in[3];
  const float* beta   = (const float*)d_in[4];
  const float* conv_w = (const float*)d_in[5];
  float* out = (float*)d_out;
  float* ws = (float*)d_ws;

  size_t off = 0;
  float* QL  = ws + off; off += (size_t)BH * MP * HD;   // landmarks
  float* QLN = ws + off; off += (size_t)BH * MP;        // 0.5||ql||^2 (pad=1e30)
  float* QN  = ws + off; off += (size_t)BH * NSEQ;      // 0.5||q||^2
  float* P   = ws + off; off += (size_t)BH * MP * MP;   // k2 (padded, zeros)
  float* VA  = ws + off; off += (size_t)BH * MP * MP;   // Newton ping
  float* VB  = ws + off; off += (size_t)BH * MP * MP;   // Newton pong
  float* T   = ws + off; off += (size_t)BH * MP * MP;   // P@V scratch
  float* Z   = ws + off; off += (size_t)BH * MP * HD;   // k1ᵀ V
  float* Y   = ws + off; off += (size_t)BH * MP * HD;   // inv Z

  landmark_kernel<<<BH, 256, 0, stream>>>(Q, w_land, gamma, beta, QL, QLN);
  qnorm_kernel<<<(BH * NSEQ + 255) / 256, 256, 0, stream>>>(Q, QN);
  k2_kernel<<<dim3(MT, MT, BH), 32, 0, stream>>>(QL, QLN, P);
  newton_init_kernel<<<BH, 256, 0, stream>>>(P, VA);

  long long sMM = (long long)MP * MP;
  float* cur = VA; float* nxt = VB;
  for (int it = 0; it < NEWTON_ITERS; ++it) {
    gemm_wmma_kernel<<<dim3(MB, MB, BH), 32, 0, stream>>>(   // T = P @ cur
        T, P, cur, P /*unused*/, MP, MP, MP, MP, MP, sMM, sMM, sMM, sMM, 0);
    gemm_wmma_kernel<<<dim3(MB, MB, BH), 32, 0, stream>>>(   // nxt = 2cur - cur@T
        nxt, cur, T, cur, MP, MP, MP, MP, MP, sMM, sMM, sMM, sMM, 1);
    float* tmp = cur; cur = nxt; nxt = tmp;
  }
  zk_kernel<<<dim3(MT, BH), 32, 0, stream>>>(Q, V, QL, QLN, QN, Z);
  gemm_wmma_kernel<<<dim3(HD / 32, MB, BH), 32, 0, stream>>>(  // Y = inv @ Z
      Y, cur, Z, cur /*unused*/, MP, HD, MP, HD, HD,
      (long long)MP * HD, sMM, (long long)MP * HD, sMM, 0);
  x_kernel<<<dim3(NT, BH), 32, 0, stream>>>(Q, V, QL, QLN, QN, Y, conv_w, out);
}